// ROIHead_45423574123085
// MI455X (gfx1250) — compile-verified
//
#include <hip/hip_runtime.h>
#include <hip/hip_bf16.h>
#include <float.h>

#define B_   4
#define NR   180
#define C_   512
#define H_   48
#define W_   48
#define F_IN 2048        // C*4
#define OUT_ 64
#define M_   (B_ * NR)   // 720
#define EPS_ 1e-5f

typedef __attribute__((ext_vector_type(2))) float v2f;
typedef __attribute__((ext_vector_type(8))) float v8f;

// ---------------------------------------------------------------------------
// Kernel 1: adaptive 2x2 ROI max-pool + ReLU.
// One block per (roi n, batch b). Lanes stride the W (x) dimension for
// coalesced row reads; waves stride channels. Wave32 shfl_xor max-reduce.
// ---------------------------------------------------------------------------
__global__ void roi_pool_relu(const float* __restrict__ feat,
                              const float* __restrict__ rois,
                              float* __restrict__ pooled) {
    const int n = blockIdx.x;
    const int b = blockIdx.y;
    const float* roi = rois + (size_t)(b * NR + n) * 4;
    const float y1 = roi[0] * 0.25f, x1 = roi[1] * 0.25f;
    const float y2 = roi[2] * 0.25f, x2 = roi[3] * 0.25f;
    int r0 = (int)truncf(y1);
    int r1 = (int)truncf(x1);
    int r2 = (int)ceilf(y2);
    int r3 = (int)ceilf(x2);
    r0 = min(max(r0, 0), H_ - 1); r2 = min(max(r2, r0), H_ - 1);
    r1 = min(max(r1, 0), W_ - 1); r3 = min(max(r3, r1), W_ - 1);
    const int hs = r2 - r0 + 1, ws = r3 - r1 + 1;
    const int eh0 = (hs + 1) >> 1, sh1 = hs >> 1;  // row region bounds
    const int ew0 = (ws + 1) >> 1, sw1 = ws >> 1;  // col region bounds
    const int lane = threadIdx.x & 31;
    const int wv   = threadIdx.x >> 5;             // 8 waves / block
    const float NEG = -FLT_MAX;

    for (int c = wv; c < C_; c += 8) {
        const float* fc = feat + ((size_t)(b * C_ + c)) * (H_ * W_);
        float m00 = NEG, m01 = NEG, m10 = NEG, m11 = NEG;
        for (int y = r0; y <= r2; ++y) {
            const int  dy = y - r0;
            const bool h0 = dy < eh0;
            const bool h1 = dy >= sh1;
            const float* fr = fc + y * W_;
            for (int x = r1 + lane; x <= r3; x += 32) {
                const int  dx = x - r1;
                const bool w0 = dx < ew0;
                const bool w1 = dx >= sw1;
                const float v = fr[x];
                if (h0 && w0) m00 = fmaxf(m00, v);
                if (h0 && w1) m01 = fmaxf(m01, v);
                if (h1 && w0) m10 = fmaxf(m10, v);
                if (h1 && w1) m11 = fmaxf(m11, v);
            }
        }
        #pragma unroll
        for (int off = 16; off > 0; off >>= 1) {
            m00 = fmaxf(m00, __shfl_xor(m00, off, 32));
            m01 = fmaxf(m01, __shfl_xor(m01, off, 32));
            m10 = fmaxf(m10, __shfl_xor(m10, off, 32));
            m11 = fmaxf(m11, __shfl_xor(m11, off, 32));
        }
        if (lane == 0) {
            float4 o;
            o.x = fmaxf(m00, 0.f); o.y = fmaxf(m01, 0.f);
            o.z = fmaxf(m10, 0.f); o.w = fmaxf(m11, 0.f);
            *(float4*)(pooled + ((size_t)(b * NR + n)) * F_IN + c * 4) = o;
        }
    }
}

// ---------------------------------------------------------------------------
// Kernel 2: BN1 statistics per roi-channel n over (B, 2048); fold gamma/beta
// into a per-n affine (s1, t1) so the GEMM can run on raw pooled data.
// ---------------------------------------------------------------------------
__global__ void bn1_stats(const float* __restrict__ pooled,
                          const float* __restrict__ gamma1,
                          const float* __restrict__ beta1,
                          float* __restrict__ s1, float* __restrict__ t1) {
    __shared__ float sb1[8], sb2[8];
    const int n = blockIdx.x, t = threadIdx.x;
    float s = 0.f, q = 0.f;
    for (int i = t; i < B_ * F_IN; i += 256) {
        const int b = i >> 11, f = i & (F_IN - 1);
        const float v = pooled[((size_t)(b * NR + n)) * F_IN + f];
        s += v; q += v * v;
    }
    #pragma unroll
    for (int off = 16; off > 0; off >>= 1) {
        s += __shfl_xor(s, off, 32);
        q += __shfl_xor(q, off, 32);
    }
    if ((t & 31) == 0) { sb1[t >> 5] = s; sb2[t >> 5] = q; }
    __syncthreads();
    if (t == 0) {
        float S = 0.f, Q = 0.f;
        #pragma unroll
        for (int i = 0; i < 8; ++i) { S += sb1[i]; Q += sb2[i]; }
        const float inv  = 1.f / (float)(B_ * F_IN);
        const float mean = S * inv;
        const float var  = Q * inv - mean * mean;
        const float sc   = rsqrtf(var + EPS_) * gamma1[n];
        s1[n] = sc;
        t1[n] = beta1[n] - mean * sc;
    }
}

// ---------------------------------------------------------------------------
// Kernel 3: row sums of W1 (needed for the folded BN1 constant term).
// ---------------------------------------------------------------------------
__global__ void wsum_k(const float* __restrict__ W1, float* __restrict__ Wsum) {
    __shared__ float sb[8];
    const int o = blockIdx.x, t = threadIdx.x;
    float s = 0.f;
    for (int f = t; f < F_IN; f += 256) s += W1[(size_t)o * F_IN + f];
    #pragma unroll
    for (int off = 16; off > 0; off >>= 1) s += __shfl_xor(s, off, 32);
    if ((t & 31) == 0) sb[t >> 5] = s;
    __syncthreads();
    if (t == 0) {
        float S = 0.f;
        #pragma unroll
        for (int i = 0; i < 8; ++i) S += sb[i];
        Wsum[o] = S;
    }
}

// ---------------------------------------------------------------------------
// Kernel 4: G = pooled @ W1^T via V_WMMA_F32_16X16X4_F32 (exact f32).
// M=720 (45 tiles), N=64 (4 waves, one 16-col tile each), K=2048.
// A frag (16x4): lane%16 = M row; VGPR pair holds K = 2*(lane/16)+{0,1}.
// B frag (4x16): lane%16 = N col (row of W1); same K striping.
// D: lane l, VGPR r -> (M = r + 8*(l/16), N = l%16).
// ---------------------------------------------------------------------------
__global__ void gemm_wmma(const float* __restrict__ A,
                          const float* __restrict__ W1,
                          float* __restrict__ G) {
    const int m0   = blockIdx.x * 16;
    const int wv   = threadIdx.x >> 5;   // 0..3 -> N tile
    const int lane = threadIdx.x & 31;
    const int row  = lane & 15;
    const int hi   = lane >> 4;          // 0/1 selects K pair
    const int n0   = wv * 16;

    const float* Ar = A  + (size_t)(m0 + row) * F_IN + 2 * hi;
    const float* Br = W1 + (size_t)(n0 + row) * F_IN + 2 * hi;

    v8f acc = {0.f, 0.f, 0.f, 0.f, 0.f, 0.f, 0.f, 0.f};
    for (int k = 0; k < F_IN; k += 4) {
        const v2f a  = *(const v2f*)(Ar + k);
        const v2f bf = *(const v2f*)(Br + k);
        acc = __builtin_amdgcn_wmma_f32_16x16x4_f32(
            /*neg_a=*/false, a, /*neg_b=*/false, bf,
            /*c_mod=*/(short)0, acc, /*reuse_a=*/false, /*reuse_b=*/false);
    }
    #pragma unroll
    for (int r = 0; r < 8; ++r)
        G[(size_t)(m0 + r + 8 * hi) * OUT_ + n0 + row] = acc[r];
}

// ---------------------------------------------------------------------------
// Kernel 5: apply folded BN1 affine + bias, then BN2 over (B,64) per n.
// One block per n; 256 threads cover all (b,o) pairs exactly.
// ---------------------------------------------------------------------------
__global__ void finalize_bn2(const float* __restrict__ G,
                             const float* __restrict__ s1,
                             const float* __restrict__ t1,
                             const float* __restrict__ Wsum,
                             const float* __restrict__ bl1,
                             const float* __restrict__ gamma2,
                             const float* __restrict__ beta2,
                             float* __restrict__ out) {
    __shared__ float sb[8];
    __shared__ float stats[2];
    const int n = blockIdx.x;
    const int t = threadIdx.x;       // 256 = 4 * 64
    const int b = t >> 6, o = t & 63;
    const float yv = s1[n] * G[((size_t)(b * NR + n)) * OUT_ + o]
                   + t1[n] * Wsum[o] + bl1[o];

    float v = yv;
    #pragma unroll
    for (int off = 16; off > 0; off >>= 1) v += __shfl_xor(v, off, 32);
    if ((t & 31) == 0) sb[t >> 5] = v;
    __syncthreads();
    if (t == 0) {
        float a = 0.f;
        #pragma unroll
        for (int i = 0; i < 8; ++i) a += sb[i];
        stats[0] = a * (1.f / 256.f);
    }
    __syncthreads();
    const float mean = stats[0];
    const float d = yv - mean;
    float v2 = d * d;
    #pragma unroll
    for (int off = 16; off > 0; off >>= 1) v2 += __shfl_xor(v2, off, 32);
    if ((t & 31) == 0) sb[t >> 5] = v2;
    __syncthreads();
    if (t == 0) {
        float a = 0.f;
        #pragma unroll
        for (int i = 0; i < 8; ++i) a += sb[i];
        stats[1] = a * (1.f / 256.f);
    }
    __syncthreads();
    const float rstd = rsqrtf(stats[1] + EPS_);
    out[((size_t)(b * NR + n)) * OUT_ + o] = d * rstd * gamma2[n] + beta2[n];
}

// ---------------------------------------------------------------------------
extern "C" void kernel_launch(void* const* d_in, const int* in_sizes, int n_in,
                              void* d_out, int out_size, void* d_ws, size_t ws_size,
                              hipStream_t stream) {
    const float* features = (const float*)d_in[0];
    const float* rois     = (const float*)d_in[1];
    const float* gamma1   = (const float*)d_in[2];
    const float* beta1    = (const float*)d_in[3];
    const float* W1       = (const float*)d_in[4];
    const float* bl1      = (const float*)d_in[5];
    const float* gamma2   = (const float*)d_in[6];
    const float* beta2    = (const float*)d_in[7];
    float* out = (float*)d_out;

    float* pooled = (float*)d_ws;                       // 720 * 2048 f32 = 5.9 MB
    float* G      = pooled + (size_t)M_ * F_IN;         // 720 * 64
    float* s1     = G + (size_t)M_ * OUT_;              // 180
    float* t1     = s1 + NR;                            // 180
    float* Wsum   = t1 + NR;                            // 64

    roi_pool_relu<<<dim3(NR, B_), 256, 0, stream>>>(features, rois, pooled);
    bn1_stats<<<NR, 256, 0, stream>>>(pooled, gamma1, beta1, s1, t1);
    wsum_k<<<OUT_, 256, 0, stream>>>(W1, Wsum);
    gemm_wmma<<<M_ / 16, 128, 0, stream>>>(pooled, W1, G);
    finalize_bn2<<<NR, 256, 0, stream>>>(G, s1, t1, Wsum, bl1, gamma2, beta2, out);
}